// Fusion_60387240182478
// MI455X (gfx1250) — compile-verified
//
#include <hip/hip_runtime.h>

// CDNA5 / gfx1250. Bandwidth-bound fusion kernel (~290 MB traffic -> ~12.5 us
// at 23.3 TB/s). Strategy:
//  - 32 points per wave-iteration; every lane owns one point (coords load,
//    out0/out2/out3 stores fully lane-parallel, no redundant loads).
//  - Streaming data (values copy, all outputs) uses non-temporal hints:
//    total footprint 290 MB > 192 MB L2, so NT avoids polluting L2;
//    tsdf/occ gathers (10 MB) stay cached.
//  - Per-point 4x4 transform via two V_WMMA_F32_16X16X4_F32 per iteration:
//    A = w2c rows 0..2 zero-padded to 16x4 (row 3 zero -> rc.w = 0),
//    B = 4x16 block of 16 homogeneous points; D rows 0..3 (lanes 0..15)
//    = (rc0, rc1, rc2, 0) = the exact float4 to store.
//  - All predicated values come from unconditional in-bounds loads + cndmask
//    selects: no EXEC-masked load branches anywhere in the hot loop.

typedef __attribute__((ext_vector_type(2))) float v2f;
typedef __attribute__((ext_vector_type(4))) float v4f;
typedef __attribute__((ext_vector_type(8))) float v8f;

namespace {
constexpr int   kD   = 64;
constexpr int   kC   = 24;
constexpr float kVsz = 0.04f;                      // VOXEL_SIZE * INTERVAL
constexpr long long kBN = 8LL * 131072LL;          // 1,048,576 points
constexpr int   kIter = (int)(kBN / 32);           // 32,768 pair-tiles of 32 points
}

__global__ void Fusion_60387240182478_kernel(
    const long long*     __restrict__ coords,   // (B*N, 4) int64
    const float*         __restrict__ values,   // (B*N, C) f32
    const float*         __restrict__ origin,   // vol_origin_partial (B,3)
    const float*         __restrict__ w2c,      // (B,4,4) f32
    const float*         __restrict__ tsdf,     // (B,D,D,D) f32
    const unsigned char* __restrict__ occ,      // (B,D,D,D) bool (1 byte)
    float*               __restrict__ out)      // BN*34 floats, concatenated
{
    const int lane   = threadIdx.x & 31;
    const int wave   = (int)((blockIdx.x * blockDim.x + threadIdx.x) >> 5);
    const int nwaves = (int)((gridDim.x * blockDim.x) >> 5);

    float* __restrict__ out0 = out;              // (BN,4)  updated_coords
    float* __restrict__ out1 = out + kBN * 4;    // (BN,24) values
    float* __restrict__ out2 = out + kBN * 28;   // (BN,1)  tsdf_target
    float* __restrict__ out3 = out + kBN * 29;   // (BN,1)  occ_target
    float* __restrict__ out4 = out + kBN * 30;   // (BN,4)  rc

    // nwaves (8192) divides kIter (32768): identical full-EXEC trip counts
    // for every wave -> WMMA always executes with EXEC all-ones.
    for (int it = wave; it < kIter; it += nwaves) {
        const long long base = (long long)it * 32;
        const long long p    = base + lane;      // this lane's point

        // ---- coords: one 32-byte (longlong4) load per lane, all 32 points distinct ----
        const longlong4 cc = ((const longlong4*)coords)[p];
        const int b = (int)cc.x;                 // uniform across the 32 points
        const int x = (int)cc.y, y = (int)cc.z, z = (int)cc.w;

        // ---- world-space position ----
        const float px = (float)x * kVsz + origin[b * 3 + 0];
        const float py = (float)y * kVsz + origin[b * 3 + 1];
        const float pz = (float)z * kVsz + origin[b * 3 + 2];

        // ---- tsdf/occ gather: unconditional load + select (no branch) ----
        const long long vox = (((long long)b * kD + x) * kD + y) * kD + z;
        const float ts = tsdf[vox];
        const float tv = occ[vox] ? ts : 1.0f;   // v_cndmask, not a branch

        // ---- per-point outputs: all 32 lanes store (non-temporal) ----
        v4f c4 = { (float)b, (float)x, (float)y, (float)z };
        __builtin_nontemporal_store(c4, (v4f*)out0 + p);
        __builtin_nontemporal_store(tv, out2 + p);
        __builtin_nontemporal_store((fabsf(tv) < 1.0f) ? 1.0f : 0.0f, out3 + p);

        // ---- values pass-through: 768 contiguous floats per iteration,
        //      6 x (32 lanes x b128); load all, then store all (pipelined) ----
        const v4f* __restrict__ vin  = (const v4f*)(values + base * kC);
        v4f*       __restrict__ vout = (v4f*)(out1 + base * kC);
        __builtin_prefetch(values + (base + (long long)nwaves * 32) * kC, 0, 1);
        v4f t[6];
#pragma unroll
        for (int i = 0; i < 6; ++i)
            t[i] = __builtin_nontemporal_load(vin + i * 32 + lane);
#pragma unroll
        for (int i = 0; i < 6; ++i)
            __builtin_nontemporal_store(t[i], vout + i * 32 + lane);

        // ---- WMMA: two 16x16x4 f32 tiles (points base..+15, base+16..+31) --
        const int m  = lane & 15;                // matrix row / point-in-tile
        const int kb = (lane >> 4) * 2;          // A K-base: 0 (lo half), 2 (hi half)
        // A-fragment: unconditional in-bounds b64 load (row m&3 of w2c),
        // then cndmask-zero rows >= 3 (row 3 zero -> rc.w = 0). No branch.
        const v2f wrow = *(const v2f*)(w2c + b * 16 + (m & 3) * 4 + kb);
        v2f af;
        af[0] = (m < 3) ? wrow[0] : 0.0f;
        af[1] = (m < 3) ? wrow[1] : 0.0f;

        // B-fragments: lane n<16 holds (px,py) of column n; lane n+16 holds
        // (pz,1) of column n. Cross-lane moves via wave32 shuffles.
        const float pz_lo = __shfl(pz, m, 32);          // pz of point (lane&15)
        const float px_hi = __shfl(px, lane | 16, 32);  // px of point (lane|16)
        const float py_hi = __shfl(py, lane | 16, 32);  // py of point (lane|16)
        v2f b0, b1;
        b0[0] = (lane < 16) ? px    : pz_lo;  b0[1] = (lane < 16) ? py    : 1.0f;
        b1[0] = (lane < 16) ? px_hi : pz;     b1[1] = (lane < 16) ? py_hi : 1.0f;

        v8f cf = {};
        v8f d0 = __builtin_amdgcn_wmma_f32_16x16x4_f32(
            false, af, false, b0, (short)0, cf, false, false);
        v8f d1 = __builtin_amdgcn_wmma_f32_16x16x4_f32(
            false, af, false, b1, (short)0, cf, false, false);

        // D rows 0..3 live in lanes 0..15: those lanes store both tiles' rc.
        if (lane < 16) {
            v4f r0 = { d0[0], d0[1], d0[2], d0[3] };
            v4f r1 = { d1[0], d1[1], d1[2], d1[3] };
            __builtin_nontemporal_store(r0, (v4f*)out4 + base + m);
            __builtin_nontemporal_store(r1, (v4f*)out4 + base + 16 + m);
        }
    }
}

extern "C" void kernel_launch(void* const* d_in, const int* in_sizes, int n_in,
                              void* d_out, int out_size, void* d_ws, size_t ws_size,
                              hipStream_t stream) {
    // setup_inputs() order:
    // 0 coords(int64), 1 values_in(f32), 2 vol_origin(f32),
    // 3 vol_origin_partial(f32), 4 world_to_aligned_camera(f32),
    // 5 tsdf_vol(f32), 6 occ_vol(bool)
    const long long*     coords = (const long long*)d_in[0];
    const float*         values = (const float*)d_in[1];
    const float*         origin = (const float*)d_in[3];   // vol_origin_partial
    const float*         w2c    = (const float*)d_in[4];
    const float*         tsdf   = (const float*)d_in[5];
    const unsigned char* occ    = (const unsigned char*)d_in[6];
    float* out = (float*)d_out;

    // 1024 blocks x 256 threads = 8192 wave32s; 32768 iterations -> exactly
    // 4 per wave, zero remainder (full-EXEC WMMA on every iteration).
    Fusion_60387240182478_kernel<<<1024, 256, 0, stream>>>(
        coords, values, origin, w2c, tsdf, occ, out);
}